// my_model1_77360950935759
// MI455X (gfx1250) — compile-verified
//
#include <hip/hip_runtime.h>
#include <math.h>

typedef float v2f __attribute__((ext_vector_type(2)));
typedef float v8f __attribute__((ext_vector_type(8)));

#define D_IN  512
#define D_HID 20
#define D_OUT 128

// D = A(16x4 f32) * B(4x16 f32) + C, full f32 precision (matches reference math)
static __device__ __forceinline__ v8f wmma_f32(v2f a, v2f b, v8f c) {
  return __builtin_amdgcn_wmma_f32_16x16x4_f32(false, a, false, b, (short)0, c,
                                               false, false);
}

__global__ void zero_f32_k(float* __restrict__ p, int n) {
  int i = blockIdx.x * blockDim.x + threadIdx.x;
  if (i < n) p[i] = 0.0f;
}

__global__ void degree_k(const long long* __restrict__ dst, float* __restrict__ deg, int E) {
  int e = blockIdx.x * blockDim.x + threadIdx.x;
  if (e < E) atomicAdd(&deg[(int)dst[e]], 1.0f);
}

// in place: deg -> rsqrt(deg + 1)   (+1 = self loop)
__global__ void dinv_k(float* __restrict__ deg, int n) {
  int i = blockIdx.x * blockDim.x + threadIdx.x;
  if (i < n) deg[i] = rsqrtf(deg[i] + 1.0f);
}

// beff[n] = bl[n] + sum_k b[k] * Wl[k,n]   (folds the GCN node bias through Wl)
__global__ void beff_k(const float* __restrict__ b1, const float* __restrict__ Wl1,
                       const float* __restrict__ bl1,
                       const float* __restrict__ b2, const float* __restrict__ Wl2,
                       const float* __restrict__ bl2,
                       float* __restrict__ beff1, float* __restrict__ beff2) {
  int t = threadIdx.x;
  if (t < D_OUT) {
    float s = bl1[t];
#pragma unroll
    for (int k = 0; k < D_HID; ++k) s += b1[k] * Wl1[k * D_OUT + t];
    beff1[t] = s;
  } else {
    int n = t - D_OUT;
    float s = bl2[n];
#pragma unroll
    for (int k = 0; k < D_HID; ++k) s += b2[k] * Wl2[k * D_OUT + n];
    beff2[n] = s;
  }
}

// One wave (32 lanes) per 16-node tile. Computes xt1 = x@W1, xt2 = x@W2 with
// v_wmma_f32_16x16x4_f32, and also writes the self-loop term agg = xt * dinv^2
// (initializes the scatter accumulators in the same pass; x is read once).
__global__ __launch_bounds__(32) void gemm1_k(const float* __restrict__ x,
                                              const float* __restrict__ W1,
                                              const float* __restrict__ W2,
                                              const float* __restrict__ dinv,
                                              float* __restrict__ xt1, float* __restrict__ xt2,
                                              float* __restrict__ agg1, float* __restrict__ agg2) {
  const int lane  = threadIdx.x & 31;
  const int base  = blockIdx.x * 16;          // node tile base (exact: 100000 = 16*6250)
  const int mrow  = lane & 15;                // A: M index / B,D: N index
  const int khalf = (lane >> 4) * 2;          // A/B: lane half selects K {0,1} vs {2,3}
  const int hi8   = (lane >> 4) * 8;          // C/D: lane half selects rows +8

  // N=20 < 32: second N-tile only has 4 valid columns -> clamped load * mask
  const float mhi  = (mrow < 4) ? 1.0f : 0.0f;
  const int   nhi  = 16 + ((mrow < 4) ? mrow : 3);

  v8f c10 = {}, c11 = {}, c20 = {}, c21 = {};
  const float* xrow = x + (size_t)(base + mrow) * D_IN + khalf;

  for (int k0 = 0; k0 < D_IN; k0 += 4) {
    float2 av = *(const float2*)(xrow + k0);   // A frag: x[m, k0+khalf .. +1]
    v2f a; a.x = av.x; a.y = av.y;
    const int kr0 = k0 + khalf, kr1 = kr0 + 1;
    v2f b;
    // W1, N-tile 0 (cols 0..15)
    b.x = W1[kr0 * D_HID + mrow]; b.y = W1[kr1 * D_HID + mrow];
    c10 = wmma_f32(a, b, c10);
    // W1, N-tile 1 (cols 16..19, masked)
    b.x = W1[kr0 * D_HID + nhi] * mhi; b.y = W1[kr1 * D_HID + nhi] * mhi;
    c11 = wmma_f32(a, b, c11);
    // W2, N-tile 0
    b.x = W2[kr0 * D_HID + mrow]; b.y = W2[kr1 * D_HID + mrow];
    c20 = wmma_f32(a, b, c20);
    // W2, N-tile 1
    b.x = W2[kr0 * D_HID + nhi] * mhi; b.y = W2[kr1 * D_HID + nhi] * mhi;
    c21 = wmma_f32(a, b, c21);
  }

#pragma unroll
  for (int r = 0; r < 8; ++r) {
    const int m = r + hi8;                    // D row for this lane half
    const float dv = dinv[base + m];
    const float dd = dv * dv;                 // self-loop coefficient
    const size_t row = (size_t)(base + m) * D_HID;
    float v;
    v = c10[r]; xt1[row + mrow] = v; agg1[row + mrow] = v * dd;
    v = c20[r]; xt2[row + mrow] = v; agg2[row + mrow] = v * dd;
    if (mrow < 4) {
      const int n = 16 + mrow;
      v = c11[r]; xt1[row + n] = v; agg1[row + n] = v * dd;
      v = c21[r]; xt2[row + n] = v; agg2[row + n] = v * dd;
    }
  }
}

// Per-edge scatter: agg[dst] += xt[src] * dinv[src]*dinv[dst] for both branches.
// xt rows are 80B -> always float4-aligned; xt/agg (8MB each) are L2-resident.
__global__ void scatter_k(const long long* __restrict__ src, const long long* __restrict__ dst,
                          const float* __restrict__ dinv,
                          const float* __restrict__ xt1, const float* __restrict__ xt2,
                          float* __restrict__ agg1, float* __restrict__ agg2, int E) {
  int e = blockIdx.x * blockDim.x + threadIdx.x;
  if (e >= E) return;
  const int s = (int)src[e];
  const int d = (int)dst[e];
  const float nrm = dinv[s] * dinv[d];
  const size_t so = (size_t)s * D_HID, dof = (size_t)d * D_HID;
  const float4* r1 = (const float4*)(xt1 + so);
  const float4* r2 = (const float4*)(xt2 + so);
  float* a1 = agg1 + dof;
  float* a2 = agg2 + dof;
#pragma unroll
  for (int i = 0; i < 5; ++i) {
    float4 v = r1[i];
    atomicAdd(a1 + 4 * i + 0, v.x * nrm);
    atomicAdd(a1 + 4 * i + 1, v.y * nrm);
    atomicAdd(a1 + 4 * i + 2, v.z * nrm);
    atomicAdd(a1 + 4 * i + 3, v.w * nrm);
  }
#pragma unroll
  for (int i = 0; i < 5; ++i) {
    float4 v = r2[i];
    atomicAdd(a2 + 4 * i + 0, v.x * nrm);
    atomicAdd(a2 + 4 * i + 1, v.y * nrm);
    atomicAdd(a2 + 4 * i + 2, v.z * nrm);
    atomicAdd(a2 + 4 * i + 3, v.w * nrm);
  }
}

// One wave per 16-node tile: m = agg @ Wl + beff (f32 WMMA, K=20 = 5 steps,
// N=128 = 8 tiles), then fused row-wise L2 normalization and store to d_out.
__global__ __launch_bounds__(32) void gemm2_norm_k(const float* __restrict__ agg,
                                                   const float* __restrict__ Wl,
                                                   const float* __restrict__ beff,
                                                   float* __restrict__ out) {
  const int lane  = threadIdx.x & 31;
  const int base  = blockIdx.x * 16;
  const int mrow  = lane & 15;
  const int khalf = (lane >> 4) * 2;
  const int hi8   = (lane >> 4) * 8;

  v8f c[8];
#pragma unroll
  for (int f = 0; f < 8; ++f) {
    const float bv = beff[f * 16 + mrow];     // bias is per-column, constant over rows
#pragma unroll
    for (int r = 0; r < 8; ++r) c[f][r] = bv;
  }

  const float* grow = agg + (size_t)(base + mrow) * D_HID + khalf;
#pragma unroll
  for (int k0 = 0; k0 < D_HID; k0 += 4) {     // 20 = 5 * 4, exact
    float2 av = *(const float2*)(grow + k0);
    v2f a; a.x = av.x; a.y = av.y;
    const int kr0 = k0 + khalf;
#pragma unroll
    for (int f = 0; f < 8; ++f) {
      v2f b;
      b.x = Wl[kr0 * D_OUT + f * 16 + mrow];
      b.y = Wl[(kr0 + 1) * D_OUT + f * 16 + mrow];
      c[f] = wmma_f32(a, b, c[f]);
    }
  }

  // Row sum-of-squares: lanes 0..15 hold row r, lanes 16..31 hold row r+8, so a
  // 4-step xor-shuffle (masks 1,2,4,8) reduces within each 16-lane half.
#pragma unroll
  for (int r = 0; r < 8; ++r) {
    float ss = 0.0f;
#pragma unroll
    for (int f = 0; f < 8; ++f) { const float v = c[f][r]; ss += v * v; }
    ss += __shfl_xor(ss, 1);
    ss += __shfl_xor(ss, 2);
    ss += __shfl_xor(ss, 4);
    ss += __shfl_xor(ss, 8);
    const float scale = 1.0f / fmaxf(sqrtf(ss), 1e-12f);
    const int m = r + hi8;
    float* orow = out + (size_t)(base + m) * D_OUT;
#pragma unroll
    for (int f = 0; f < 8; ++f) orow[f * 16 + mrow] = c[f][r] * scale;
  }
}

extern "C" void kernel_launch(void* const* d_in, const int* in_sizes, int n_in,
                              void* d_out, int out_size, void* d_ws, size_t ws_size,
                              hipStream_t stream) {
  const float*     x   = (const float*)d_in[0];
  const long long* ei  = (const long long*)d_in[1];   // int64 [2, E]
  const float*     W1  = (const float*)d_in[2];
  const float*     b1  = (const float*)d_in[3];
  const float*     W2  = (const float*)d_in[4];
  const float*     b2  = (const float*)d_in[5];
  const float*     Wl1 = (const float*)d_in[6];
  const float*     bl1 = (const float*)d_in[7];
  const float*     Wl2 = (const float*)d_in[8];
  const float*     bl2 = (const float*)d_in[9];
  float* out = (float*)d_out;

  const int N = in_sizes[0] / D_IN;        // 100000
  const int E = in_sizes[1] / 2;           // 3200000
  const long long* src = ei;
  const long long* dst = ei + E;

  // Workspace layout (floats); all tile bases 16B-aligned.
  float* ws = (float*)d_ws;
  size_t o = 0;
  float* deg   = ws + o; o += (size_t)N;       // becomes dinv in place
  float* beff1 = ws + o; o += 128;
  float* beff2 = ws + o; o += 128;
  float* xt1   = ws + o; o += (size_t)N * D_HID;
  float* xt2   = ws + o; o += (size_t)N * D_HID;
  float* agg1  = ws + o; o += (size_t)N * D_HID;
  float* agg2  = ws + o; o += (size_t)N * D_HID;
  (void)ws_size; (void)out_size; (void)n_in;

  const int tiles = N / 16;                // 6250 exact -> EXEC always all-ones

  zero_f32_k<<<(N + 255) / 256, 256, 0, stream>>>(deg, N);
  degree_k<<<(E + 255) / 256, 256, 0, stream>>>(dst, deg, E);
  dinv_k<<<(N + 255) / 256, 256, 0, stream>>>(deg, N);
  beff_k<<<1, 256, 0, stream>>>(b1, Wl1, bl1, b2, Wl2, bl2, beff1, beff2);
  gemm1_k<<<tiles, 32, 0, stream>>>(x, W1, W2, deg, xt1, xt2, agg1, agg2);
  scatter_k<<<(E + 255) / 256, 256, 0, stream>>>(src, dst, deg, xt1, xt2, agg1, agg2, E);
  gemm2_norm_k<<<tiles, 32, 0, stream>>>(agg1, Wl1, beff1, out);
  gemm2_norm_k<<<tiles, 32, 0, stream>>>(agg2, Wl2, beff2, out + (size_t)N * D_OUT);
}